// LDAMLoss_6305011990635
// MI455X (gfx1250) — compile-verified
//
#include <hip/hip_runtime.h>
#include <stdint.h>

// LDAM loss, fused: per-row margin-adjusted log-softmax NLL + mean.
// B=32768 rows, C=1000 cols, S=30.
// Memory floor: 131MB @ 23.3 TB/s ~ 5.6us. VALU trimmed to ~4 ops/element
// (max; fma+exp2+add) by working in base-2 and applying the single-element
// margin adjustment as a post-correction to the exp-sum instead of a
// per-element compare/select.
// CDNA5 path: async global->LDS b128 staging (ASYNCcnt), double-buffered per
// wave32, s_wait_asynccnt software pipeline, wave32 shuffle reductions.

#define LDAM_B 32768
#define LDAM_C 1000

#define WAVES_PER_BLOCK 8
#define BLOCKS 1024             // 8192 waves -> 4 rows/wave
#define ROW_F4 250              // 1000 floats = 250 float4 chunks
#define BUF_F32 1024            // padded row buffer (floats)
#define BUF_BYTES (BUF_F32 * 4)

// S * log2(e) and ln(2)
#define LDAM_S2  43.28085122666891f   // 30 * 1.4426950408889634
#define LDAM_LN2 0.6931471805599453f

__device__ __forceinline__ void async_prefetch_row(const float* __restrict__ logits,
                                                   int row, unsigned buf_byte_off, int lane) {
    const float* gr = logits + (size_t)row * LDAM_C;
#pragma unroll
    for (int i = 0; i < 8; ++i) {
        int c = i * 32 + lane;          // float4 chunk index, 0..255
        if (c < ROW_F4) {
            unsigned laddr = buf_byte_off + (unsigned)c * 16u;
            const float* ga = gr + c * 4;
            // async DMA: 16B global -> LDS, tracked by ASYNCcnt
            asm volatile("global_load_async_to_lds_b128 %0, %1, off"
                         :: "v"(laddr), "v"(ga) : "memory");
        }
    }
}

__global__ void __launch_bounds__(256)
ldam_rows_kernel(const float* __restrict__ logits,
                 const float* __restrict__ margins,
                 const int*   __restrict__ targets,
                 float*       __restrict__ block_partial) {
    __shared__ __align__(16) float smem[WAVES_PER_BLOCK * 2 * BUF_F32];
    __shared__ float red[WAVES_PER_BLOCK];

    const int lane = threadIdx.x & 31;
    const int warp = threadIdx.x >> 5;
    const int wave_id = blockIdx.x * WAVES_PER_BLOCK + warp;
    const int stride  = gridDim.x * WAVES_PER_BLOCK;

    // LDS byte offset of this wave's two buffers (generic->LDS offset = low 32 bits)
    const unsigned lds_base = (unsigned)(uintptr_t)(void*)smem;
    unsigned buf_off[2];
    buf_off[0] = lds_base + (unsigned)(warp * 2 + 0) * BUF_BYTES;
    buf_off[1] = lds_base + (unsigned)(warp * 2 + 1) * BUF_BYTES;
    float* bufp[2];
    bufp[0] = smem + (warp * 2 + 0) * BUF_F32;
    bufp[1] = smem + (warp * 2 + 1) * BUF_F32;

    int p = 0;
    if (wave_id < LDAM_B) async_prefetch_row(logits, wave_id, buf_off[0], lane);

    float acc = 0.0f;   // per-wave sum of nll over its rows

    for (int r = wave_id; r < LDAM_B; r += stride) {
        const int nxt = r + stride;
        if (nxt < LDAM_B) {
            async_prefetch_row(logits, nxt, buf_off[p ^ 1], lane);
            // async loads complete in order: <=8 outstanding => current row landed
            asm volatile("s_wait_asynccnt 8" ::: "memory");
        } else {
            asm volatile("s_wait_asynccnt 0" ::: "memory");
        }

        const int   t  = targets[r];
        const float mg = margins[t];
        const float4* bv = reinterpret_cast<const float4*>(bufp[p]);

        // Pass 1: raw row values into registers + running max (1 VALU/elt).
        // Margin handled later as a correction (it only lowers one element,
        // so the raw max remains a safe softmax shift).
        float vals[32];
        float m0 = -INFINITY, m1 = -INFINITY;
#pragma unroll
        for (int i = 0; i < 8; ++i) {
            const int c = i * 32 + lane;          // 0..255, LDS read in-bounds
            const float4 v = bv[c];
            const bool valid = (c < ROW_F4);
            const float a0 = valid ? v.x : -INFINITY;
            const float a1 = valid ? v.y : -INFINITY;
            const float a2 = valid ? v.z : -INFINITY;
            const float a3 = valid ? v.w : -INFINITY;
            vals[i * 4 + 0] = a0;
            vals[i * 4 + 1] = a1;
            vals[i * 4 + 2] = a2;
            vals[i * 4 + 3] = a3;
            m0 = fmaxf(m0, fmaxf(a0, a1));
            m1 = fmaxf(m1, fmaxf(a2, a3));
        }
        float xmax = fmaxf(m0, m1);
#pragma unroll
        for (int o = 16; o > 0; o >>= 1)
            xmax = fmaxf(xmax, __shfl_xor(xmax, o, 32));
        const float bmax = LDAM_S2 * xmax;        // base-2 shift

        // Pass 2: sum exp2(S2*x - bmax): fma + v_exp + add per element,
        // 4 accumulators to let trans-unit exp2 co-execute with the adds.
        float s0 = 0.0f, s1 = 0.0f, s2 = 0.0f, s3 = 0.0f;
#pragma unroll
        for (int i = 0; i < 8; ++i) {
            s0 += __builtin_amdgcn_exp2f(fmaf(LDAM_S2, vals[i * 4 + 0], -bmax));
            s1 += __builtin_amdgcn_exp2f(fmaf(LDAM_S2, vals[i * 4 + 1], -bmax));
            s2 += __builtin_amdgcn_exp2f(fmaf(LDAM_S2, vals[i * 4 + 2], -bmax));
            s3 += __builtin_amdgcn_exp2f(fmaf(LDAM_S2, vals[i * 4 + 3], -bmax));
        }
        float ssum = (s0 + s1) + (s2 + s3);
#pragma unroll
        for (int o = 16; o > 0; o >>= 1)
            ssum += __shfl_xor(ssum, o, 32);

        // Margin correction at the target column (uniform scalar load, L2-hot):
        // replace exp2(bt_raw) term with exp2(bt_raw - S2*m).
        const float xt      = logits[(size_t)r * LDAM_C + t];
        const float bt_raw  = fmaf(LDAM_S2, xt, -bmax);       // rel. to bmax
        const float bt_adj  = bt_raw - LDAM_S2 * mg;
        const float scorr   = ssum - __builtin_amdgcn_exp2f(bt_raw)
                                   + __builtin_amdgcn_exp2f(bt_adj);
        // nll = ln2 * (log2(sum_rel) - bt_adj)   (bmax cancels)
        acc += LDAM_LN2 * (__builtin_amdgcn_logf(scorr) - bt_adj);

        p ^= 1;
    }

    // Deterministic per-block partial: fixed-order 8-wave combine.
    if (lane == 0) red[warp] = acc;
    __syncthreads();
    if (threadIdx.x == 0) {
        float s = 0.0f;
#pragma unroll
        for (int w = 0; w < WAVES_PER_BLOCK; ++w) s += red[w];
        block_partial[blockIdx.x] = s;
    }
}

__global__ void __launch_bounds__(256)
ldam_final_kernel(const float* __restrict__ block_partial, float* __restrict__ out) {
    __shared__ float red[256];
    const int tid = threadIdx.x;
    float s = (block_partial[tid] + block_partial[tid + 256])
            + (block_partial[tid + 512] + block_partial[tid + 768]);
    red[tid] = s;
    __syncthreads();
#pragma unroll
    for (int o = 128; o > 0; o >>= 1) {
        if (tid < o) red[tid] += red[tid + o];
        __syncthreads();
    }
    if (tid == 0)
        out[0] = red[0] * (1.0f / (float)LDAM_B);
}

extern "C" void kernel_launch(void* const* d_in, const int* in_sizes, int n_in,
                              void* d_out, int out_size, void* d_ws, size_t ws_size,
                              hipStream_t stream) {
    const float* logits  = (const float*)d_in[0];   // [32768, 1000] f32
    const float* margins = (const float*)d_in[1];   // [1000] f32
    const int*   targets = (const int*)d_in[2];     // [32768] int32
    float* out = (float*)d_out;                     // scalar mean (f32)
    float* partials = (float*)d_ws;                 // >= 1024 floats

    ldam_rows_kernel<<<dim3(BLOCKS), dim3(256), 0, stream>>>(logits, margins, targets, partials);
    ldam_final_kernel<<<dim3(1), dim3(256), 0, stream>>>(partials, out);
}